// GroupLinear_59030030516596
// MI455X (gfx1250) — compile-verified
//
#include <hip/hip_runtime.h>

typedef float v2f __attribute__((ext_vector_type(2)));
typedef float v8f __attribute__((ext_vector_type(8)));

#define T_TOK 65536
#define NGRP  64
#define DIN   1024
#define DOUT  1024

#define BM  128
#define BN  128
#define BK  32
#define BKP 36              // padded row stride (floats): conflict-free, 16B-aligned rows
#define KT  (DIN / BK)      // 32 k-tiles

__global__ __launch_bounds__(256) void grouped_gemm_f32_wmma(
    const float* __restrict__ x,      // [T, DIN]
    const float* __restrict__ w,      // [G, DOUT, DIN]
    const int*   __restrict__ offs,   // [G] cumulative end offsets
    float*       __restrict__ out)    // [T, DOUT]
{
    __shared__ float As[2][BM][BKP];
    __shared__ float Bs[2][BN][BKP];

    const int g      = blockIdx.z;
    const int gEnd   = offs[g];
    const int gStart = (g == 0) ? 0 : offs[g - 1];
    const int mBase  = blockIdx.y * BM;
    if (mBase >= gEnd - gStart) return;           // tile beyond this group's rows

    const int rowBase = gStart + mBase;
    const int colBase = blockIdx.x * BN;

    const int tid  = threadIdx.x;
    const int lane = tid & 31;        // wave32
    const int wave = tid >> 5;        // 0..7
    const int wm   = wave >> 1;       // 0..3 : 32-row band
    const int wn   = wave & 1;        // 0..1 : 64-col half
    const int half = lane >> 4;       // 0/1  : K-pair select / M-upper-half for C
    const int l16  = lane & 15;       // M (A) / N (B,C) index

    const float* wg = w + (size_t)g * DOUT * DIN;

    // global->LDS async pattern: 8 threads per row, b128 along K, 4 row-passes
    const int kc = (tid & 7) * 4;     // float index 0,4,...,28
    const int rr = tid >> 3;          // 0..31

    v8f acc[2][4];
    #pragma unroll
    for (int i = 0; i < 2; ++i)
        #pragma unroll
        for (int j = 0; j < 4; ++j)
            acc[i][j] = {};

    // Issue 8 async GLOBAL->LDS b128 descriptors for k-tile kt into buffer `buf`.
    // GVS mode: uniform SGPR base + per-lane 32-bit byte offset. LDS address is
    // the low 32 bits of the generic pointer (ISA: flat LDS addr = {SHARED_BASE, off32}).
    auto issue_async = [&](int kt, int buf) {
        const int kB = kt * BK;
        #pragma unroll
        for (int p = 0; p < 4; ++p) {
            const int row = p * 32 + rr;
            int gr = rowBase + row;
            gr = (gr < T_TOK) ? gr : (T_TOK - 1);   // clamp keeps issue count uniform
            const unsigned aoff = ((unsigned)gr * DIN + (unsigned)(kB + kc)) * 4u;
            const unsigned boff = ((unsigned)(colBase + row) * DIN + (unsigned)(kB + kc)) * 4u;
            const unsigned lA = (unsigned)(size_t)(const void*)&As[buf][row][kc];
            const unsigned lB = (unsigned)(size_t)(const void*)&Bs[buf][row][kc];
            asm volatile("global_load_async_to_lds_b128 %0, %1, %2"
                         :: "v"(lA), "v"(aoff), "s"(x) : "memory");
            asm volatile("global_load_async_to_lds_b128 %0, %1, %2"
                         :: "v"(lB), "v"(boff), "s"(wg) : "memory");
        }
    };

    auto compute = [&](int buf) {
        #pragma unroll
        for (int kk = 0; kk < BK; kk += 4) {
            v2f af[2], bf[4];
            #pragma unroll
            for (int i = 0; i < 2; ++i)
                af[i] = *(const v2f*)&As[buf][wm * 32 + i * 16 + l16][kk + half * 2];
            #pragma unroll
            for (int j = 0; j < 4; ++j)
                bf[j] = *(const v2f*)&Bs[buf][wn * 64 + j * 16 + l16][kk + half * 2];
            #pragma unroll
            for (int i = 0; i < 2; ++i)
                #pragma unroll
                for (int j = 0; j < 4; ++j)
                    acc[i][j] = __builtin_amdgcn_wmma_f32_16x16x4_f32(
                        false, af[i], false, bf[j],
                        (short)0, acc[i][j], false, false);
        }
    };

    // prologue: stage tile 0 asynchronously
    issue_async(0, 0);

    for (int kt = 0; kt < KT; ++kt) {
        const int buf = kt & 1;

        if (kt + 1 < KT) {
            issue_async(kt + 1, buf ^ 1);           // prefetch next tile into other buffer
            // warm GL2 for the tile after next (global_prefetch_b8)
            if (kt + 2 < KT) {
                const int kB2 = (kt + 2) * BK;
                const int gr  = rowBase + rr;
                if (gr < T_TOK)
                    __builtin_prefetch(&x[(size_t)gr * DIN + kB2 + kc], 0, 1);
                __builtin_prefetch(&wg[(size_t)(colBase + rr) * DIN + kB2 + kc], 0, 1);
            }
            // in-order completion: once <=8 remain, this tile's 8 ops are done
            asm volatile("s_wait_asynccnt 8" ::: "memory");
        } else {
            asm volatile("s_wait_asynccnt 0" ::: "memory");
        }

        __syncthreads();        // publish this buffer's LDS writes to all waves
        compute(buf);
        __syncthreads();        // all waves done reading before next issue overwrites
    }

    // writeback: C layout — lane l16 = column, VGPR r = row (r) or (8+r) by lane half
    #pragma unroll
    for (int i = 0; i < 2; ++i) {
        #pragma unroll
        for (int j = 0; j < 4; ++j) {
            const int col = colBase + wn * 64 + j * 16 + l16;
            #pragma unroll
            for (int r = 0; r < 8; ++r) {
                const int row = rowBase + wm * 32 + i * 16 + half * 8 + r;
                if (row < gEnd)
                    out[(size_t)row * DOUT + col] = acc[i][j][r];
            }
        }
    }
}

extern "C" void kernel_launch(void* const* d_in, const int* in_sizes, int n_in,
                              void* d_out, int out_size, void* d_ws, size_t ws_size,
                              hipStream_t stream) {
    const float* x    = (const float*)d_in[0];   // [T, DIN] f32
    const float* w    = (const float*)d_in[1];   // [G, DOUT, DIN] f32
    const int*   offs = (const int*)d_in[2];     // [G] i32 cumulative ends
    float*       out  = (float*)d_out;           // [T, DOUT] f32

    dim3 grid(DOUT / BN, (T_TOK + BM - 1) / BM, NGRP);   // 8 x 512 x 64
    dim3 block(256);
    grouped_gemm_f32_wmma<<<grid, block, 0, stream>>>(x, w, offs, out);
}